// graph_autoencoder_79680233276336
// MI455X (gfx1250) — compile-verified
//
#include <hip/hip_runtime.h>
#include <hip/hip_bf16.h>
#include <stdint.h>

// Problem constants (match reference)
#define N_NODES 50000
#define N_EDGES 800000
#define FEATSZ  128
#define HIDSZ   128
#define LATSZ   64
#define NGRAPH  256
#define EN      (N_EDGES + N_NODES)   // edges + self loops

typedef __attribute__((ext_vector_type(2))) float v2f;
typedef __attribute__((ext_vector_type(8))) float v8f;

// ---------------- elementwise helpers ----------------
__global__ void k_fill(float* __restrict__ p, float v, int n) {
  int i = blockIdx.x * blockDim.x + threadIdx.x;
  if (i < n) p[i] = v;
}

// out[r][f] = bias[f]; feat must be power of two (128 here)
__global__ void k_fill_bias(float* __restrict__ out, const float* __restrict__ bias,
                            int feat, int total) {
  int i = blockIdx.x * blockDim.x + threadIdx.x;
  if (i < total) out[i] = bias[i & (feat - 1)];
}

// ---------------- GCN normalization ----------------
__global__ void k_deg_accum(const int* __restrict__ col, float* __restrict__ deg) {
  int e = blockIdx.x * blockDim.x + threadIdx.x;
  if (e < N_EDGES) atomicAdd(&deg[col[e]], 1.0f);
}

__global__ void k_rsqrt(float* __restrict__ p, int n) {
  int i = blockIdx.x * blockDim.x + threadIdx.x;
  if (i < n) p[i] = rsqrtf(p[i]);   // deg >= 1 always (self loop)
}

__global__ void k_norm(const int* __restrict__ row, const int* __restrict__ col,
                       const float* __restrict__ dinv, float* __restrict__ norm) {
  int e = blockIdx.x * blockDim.x + threadIdx.x;
  if (e >= EN) return;
  int r, c;
  if (e < N_EDGES) { r = row[e]; c = col[e]; }
  else             { r = c = e - N_EDGES; }       // self loops
  norm[e] = dinv[r] * dinv[c];
}

// ---- WMMA f32 GEMM: C[M,Nc] = gather(relu?(A))[M,K] @ W[K,Nc] (+bias)(+relu) ----
// Block = 8 waves (256 threads). The block's K x 64 W-slab is staged once into
// LDS in PAIR-INTERLEAVED order: sW[(k/2)*128 + c*2 + (k&1)], so each lane's
// B fragment {W[k][c], W[k+1][c]} is one contiguous ds_load_b64 directly into
// the WMMA operand register pair (no v_mov shuffles). Each wave computes a
// 16(M) x 64(N) strip: the A fragment (one global_load_b64) is reused across 4
// V_WMMA_F32_16X16X4_F32 per k-step. Fusion flags are template params so no
// runtime v_cndmask appears in the hot loop. Full f32 precision.
// Requires: Nc % 64 == 0, K % 4 == 0, M % 16 == 0. EXEC all-1s inside waves.
template <int INR, int OUTR, bool IDX, bool BIAS>
__global__ __launch_bounds__(256)
void k_gemm_wmma(const float* __restrict__ A, const float* __restrict__ W,
                 const float* __restrict__ bias, float* __restrict__ C,
                 const int* __restrict__ rowidx, int M, int K, int Nc) {
  extern __shared__ float sW[];                       // [K/2][64][2] interleaved
  const int lane = threadIdx.x;                       // 0..31 (wave32)
  const int tid  = threadIdx.y * 32 + lane;           // 0..255
  const int nb   = blockIdx.x * 64;                   // column base of this block

  // cooperative stage of W[:, nb:nb+64] into LDS, pair-interleaved
  for (int i = tid; i < K * 64; i += 256) {
    int k = i >> 6, c = i & 63;
    sW[(k >> 1) * 128 + c * 2 + (k & 1)] = W[(size_t)k * Nc + nb + c];
  }
  __syncthreads();

  const int mt = blockIdx.y * blockDim.y + threadIdx.y;
  if (mt * 16 < M) {                                  // wave-uniform guard
    const int m0 = mt * 16;
    // ISA layout, 32-bit A 16x4: lanes 0-15 -> K=0/1, lanes 16-31 -> K=2/3
    const int mh   = lane & 15;
    const int kh   = lane >> 4;                       // 0 or 1
    const int koff = kh << 1;

    const int am   = m0 + mh;
    const int asrc = IDX ? rowidx[am] : am;           // fused row-gather (z[batch])
    const float* __restrict__ ap = A + (size_t)asrc * K;

    v8f acc0 = {0.f,0.f,0.f,0.f,0.f,0.f,0.f,0.f};
    v8f acc1 = acc0, acc2 = acc0, acc3 = acc0;

    for (int k0 = 0; k0 < K; k0 += 4) {
      v2f a = *(const v2f*)&ap[k0 + koff];            // global_load_b64
      if (INR) { a.x = fmaxf(a.x, 0.0f); a.y = fmaxf(a.y, 0.0f); }
      const int pb = ((k0 >> 1) + kh) * 128 + mh * 2; // pair row for this lane
      v2f b0 = *(const v2f*)&sW[pb];                  // ds_load_b64 each
      v2f b1 = *(const v2f*)&sW[pb + 32];
      v2f b2 = *(const v2f*)&sW[pb + 64];
      v2f b3 = *(const v2f*)&sW[pb + 96];
      acc0 = __builtin_amdgcn_wmma_f32_16x16x4_f32(false, a, false, b0, (short)0, acc0, false, false);
      acc1 = __builtin_amdgcn_wmma_f32_16x16x4_f32(false, a, false, b1, (short)0, acc1, false, false);
      acc2 = __builtin_amdgcn_wmma_f32_16x16x4_f32(false, a, false, b2, (short)0, acc2, false, false);
      acc3 = __builtin_amdgcn_wmma_f32_16x16x4_f32(false, a, false, b3, (short)0, acc3, false, false);
    }

    // C/D layout: lane<16 -> M = 0..7 in v0..v7; lane>=16 -> M = 8..15; N = lane&15
    const int crow0 = m0 + (kh << 3);
    v8f accs[4] = {acc0, acc1, acc2, acc3};
#pragma unroll
    for (int t = 0; t < 4; ++t) {
      const int ccol = nb + t * 16 + mh;
      const float bv = BIAS ? bias[ccol] : 0.0f;
#pragma unroll
      for (int v = 0; v < 8; ++v) {
        float r = accs[t][v] + bv;
        if (OUTR) r = fmaxf(r, 0.0f);
        C[(size_t)(crow0 + v) * Nc + ccol] = r;
      }
    }
  }
}

// ---------------- message scatter: out[col[e]] += norm[e] * hW[row[e]] ----------------
// One wave per edge (incl. self loops); each atomic instruction covers a
// contiguous 128B span across the wave. All operands L2-resident (<<192MB).
__global__ __launch_bounds__(256)
void k_scatter(const float* __restrict__ hW, const float* __restrict__ norm,
               const int* __restrict__ row, const int* __restrict__ col,
               float* __restrict__ out, int feat) {
  int wave = blockIdx.x * (blockDim.x >> 5) + (threadIdx.x >> 5);
  int lane = threadIdx.x & 31;
  if (wave >= EN) return;
  int r, c;
  if (wave < N_EDGES) { r = row[wave]; c = col[wave]; }
  else                { r = c = wave - N_EDGES; }
  float nm = norm[wave];
  const float* __restrict__ src = hW + (size_t)r * feat;
  float* __restrict__ dst = out + (size_t)c * feat;
  for (int f = lane; f < feat; f += 32)
    atomicAdd(&dst[f], nm * src[f]);
}

// ---------------- global_add_pool with fused ReLU: pooled[batch[n]] += relu(h[n]) ----
__global__ __launch_bounds__(256)
void k_pool_relu(const float* __restrict__ h, const int* __restrict__ batch,
                 float* __restrict__ pooled) {
  int wave = blockIdx.x * (blockDim.x >> 5) + (threadIdx.x >> 5);
  int lane = threadIdx.x & 31;
  if (wave >= N_NODES) return;
  int g = batch[wave];
  const float* __restrict__ src = h + (size_t)wave * HIDSZ;
  float* __restrict__ dst = pooled + (size_t)g * HIDSZ;
  for (int f = lane; f < HIDSZ; f += 32)
    atomicAdd(&dst[f], fmaxf(src[f], 0.0f));
}

extern "C" void kernel_launch(void* const* d_in, const int* in_sizes, int n_in,
                              void* d_out, int out_size, void* d_ws, size_t ws_size,
                              hipStream_t stream) {
  const float* x     = (const float*)d_in[0];
  const int*   ei    = (const int*)  d_in[1];   // [2, E]
  const int*   batch = (const int*)  d_in[2];   // [N]
  const float* W1  = (const float*)d_in[3];
  const float* b1  = (const float*)d_in[4];
  const float* W2  = (const float*)d_in[5];
  const float* b2  = (const float*)d_in[6];
  const float* fcW = (const float*)d_in[7];
  const float* fcb = (const float*)d_in[8];
  const float* dW  = (const float*)d_in[9];
  const float* db  = (const float*)d_in[10];
  const float* W3  = (const float*)d_in[11];
  const float* b3  = (const float*)d_in[12];

  const int* row = ei;             // edge_index[0]
  const int* col = ei + N_EDGES;   // edge_index[1]

  float* out    = (float*)d_out;                        // recon [N, FEAT]
  float* latent = out + (size_t)N_NODES * FEATSZ;       // latent [G, LAT]

  // workspace carve-up (~55 MB)
  float* w = (float*)d_ws;
  float* dinv   = w;  w += N_NODES;
  float* norm   = w;  w += EN;
  float* bufA   = w;  w += (size_t)N_NODES * HIDSZ;     // h@W staging
  float* bufB   = w;  w += (size_t)N_NODES * HIDSZ;     // activations (pre-relu)
  float* pooled = w;  w += NGRAPH * HIDSZ;
  float* zbuf   = w;  w += NGRAPH * HIDSZ;

  const int T = 256;
  dim3 gblk(32, 8);                                     // 8 waves / block
  auto ggrid = [](int M, int Nc) { return dim3(Nc / 64, ((M / 16) + 7) / 8); };
  auto gshm  = [](int K) { return (size_t)K * 64 * sizeof(float); };

  // --- GCN normalization (shared by all convs) ---
  k_fill     <<<(N_NODES + T - 1) / T, T, 0, stream>>>(dinv, 1.0f, N_NODES); // self-loop deg
  k_deg_accum<<<(N_EDGES + T - 1) / T, T, 0, stream>>>(col, dinv);
  k_rsqrt    <<<(N_NODES + T - 1) / T, T, 0, stream>>>(dinv, N_NODES);
  k_norm     <<<(EN + T - 1) / T, T, 0, stream>>>(row, col, dinv, norm);

  // --- encoder layer 1: h1 = gcn(x, W1, b1) (pre-relu; relu fused downstream) ---
  k_gemm_wmma<0,0,false,false><<<ggrid(N_NODES, HIDSZ), gblk, gshm(FEATSZ), stream>>>(
      x, W1, nullptr, bufA, nullptr, N_NODES, FEATSZ, HIDSZ);
  k_fill_bias<<<(N_NODES * HIDSZ + T - 1) / T, T, 0, stream>>>(bufB, b1, HIDSZ, N_NODES * HIDSZ);
  k_scatter  <<<(EN + 7) / 8, T, 0, stream>>>(bufA, norm, row, col, bufB, HIDSZ);

  // --- encoder layer 2: h2 = gcn(relu(h1), W2, b2); relu(h1) fused into A reads ---
  k_gemm_wmma<1,0,false,false><<<ggrid(N_NODES, HIDSZ), gblk, gshm(HIDSZ), stream>>>(
      bufB, W2, nullptr, bufA, nullptr, N_NODES, HIDSZ, HIDSZ);
  k_fill_bias<<<(N_NODES * HIDSZ + T - 1) / T, T, 0, stream>>>(bufB, b2, HIDSZ, N_NODES * HIDSZ);
  k_scatter  <<<(EN + 7) / 8, T, 0, stream>>>(bufA, norm, row, col, bufB, HIDSZ);

  // --- pool (relu(h2) fused) + fc: latent = relu(pooled @ fcW + fcb) ---
  k_fill<<<(NGRAPH * HIDSZ + T - 1) / T, T, 0, stream>>>(pooled, 0.0f, NGRAPH * HIDSZ);
  k_pool_relu<<<(N_NODES + 7) / 8, T, 0, stream>>>(bufB, batch, pooled);
  k_gemm_wmma<0,1,false,true><<<ggrid(NGRAPH, LATSZ), gblk, gshm(HIDSZ), stream>>>(
      pooled, fcW, fcb, latent, nullptr, NGRAPH, HIDSZ, LATSZ);

  // --- decoder: z = relu(latent @ decW + decb) ---
  k_gemm_wmma<0,1,false,true><<<ggrid(NGRAPH, HIDSZ), gblk, gshm(LATSZ), stream>>>(
      latent, dW, db, zbuf, nullptr, NGRAPH, LATSZ, HIDSZ);

  // --- layer 3: recon = gcn(z[batch], W3, b3); broadcast fused into GEMM A-gather ---
  k_gemm_wmma<0,0,true,false><<<ggrid(N_NODES, FEATSZ), gblk, gshm(HIDSZ), stream>>>(
      zbuf, W3, nullptr, bufA, batch, N_NODES, HIDSZ, FEATSZ);
  k_fill_bias<<<(N_NODES * FEATSZ + T - 1) / T, T, 0, stream>>>(out, b3, FEATSZ,
                                                                N_NODES * FEATSZ);
  k_scatter  <<<(EN + 7) / 8, T, 0, stream>>>(bufA, norm, row, col, out, FEATSZ);
}